// LowRank_47012712022298
// MI455X (gfx1250) — compile-verified
//
#include <hip/hip_runtime.h>
#include <math.h>

typedef __attribute__((ext_vector_type(2))) float v2f;
typedef __attribute__((ext_vector_type(8))) float v8f;
typedef int vi4 __attribute__((vector_size(16)));   // matches async-LDS builtin param type

#define B_   32
#define N_   1024
#define E_   1024
#define H_   8
#define D_   128
#define MID_ 64
#define SPLIT_ 8      // workgroups per batch element in the key-streaming kernel
#define VS_    16     // workgroups per batch element in the value2 reduction

// workspace layout (float offsets)
#define OFF_Q     0
#define OFF_QK    (OFF_Q    + B_*E_)      // qk[b][h][e]
#define OFF_QBK   (OFF_QK   + B_*H_*E_)   // qbk[b][h]
#define OFF_SKEY  (OFF_QBK  + B_*H_)      // s_key[b][h][e]  (atomically accumulated)
#define OFF_SUMR  (OFF_SKEY + B_*H_*E_)   // sum of raw scores [b][h]
#define OFF_VSUM  (OFF_SUMR + B_*H_)      // masked sum of value2 [b][e]
#define WS_TOTAL  (OFF_VSUM + B_*E_)
#define ZERO_BEGIN OFF_SKEY
#define ZERO_COUNT (WS_TOTAL - OFF_SKEY)

// CDNA5 async global->LDS path (ASYNCcnt-tracked, no VGPR staging)
#if defined(__AMDGCN__) && __has_builtin(__builtin_amdgcn_global_load_async_to_lds_b128) && \
    __has_builtin(__builtin_amdgcn_s_wait_asynccnt)
#define ASYNC_TILES 1
#else
#define ASYNC_TILES 0
#endif

__global__ void zero_ws(float* ws) {
    int i = blockIdx.x * 256 + threadIdx.x;
    if (i < ZERO_COUNT) ws[ZERO_BEGIN + i] = 0.f;
}

// Q[b][c] = query[b] . Wq[c] + bq[c]
__global__ void q_proj(const float* __restrict__ query, const float* __restrict__ Wq,
                       const float* __restrict__ bq, float* __restrict__ Q) {
    int o = blockIdx.x * 256 + threadIdx.x;       // B_*E_ outputs
    int b = o >> 10, c = o & 1023;
    const float* q = query + b * E_;
    const float* w = Wq + (size_t)c * E_;
    float acc = 0.f;
    for (int e = 0; e < E_; ++e) acc += q[e] * w[e];
    Q[o] = acc + bq[c];
}

// qk[b][h][e] = sum_d Wk[h*D+d][e] * Q[b][h*D+d]
__global__ void qk_proj(const float* __restrict__ Q, const float* __restrict__ Wk,
                        float* __restrict__ qk) {
    int o = blockIdx.x * 256 + threadIdx.x;       // B_*H_*E_ outputs
    int e = o & 1023, h = (o >> 10) & 7, b = o >> 13;
    const float* qr = Q + b * E_ + h * D_;
    float acc = 0.f;
    for (int d = 0; d < D_; ++d) acc += Wk[(size_t)(h * D_ + d) * E_ + e] * qr[d];
    qk[o] = acc;
}

// qbk[b][h] = Q[b,h,:] . bk[h*D:+D]
__global__ void qbk_proj(const float* __restrict__ Q, const float* __restrict__ bk,
                         float* __restrict__ qbk) {
    int t = threadIdx.x;                           // 256 = B_*H_
    int b = t >> 3, h = t & 7;
    const float* qr  = Q + b * E_ + h * D_;
    const float* bkr = bk + h * D_;
    float acc = 0.f;
    for (int d = 0; d < D_; ++d) acc += qr[d] * bkr[d];
    qbk[t] = acc;
}

// --------- main streaming kernel: one pass over `key` (128 MB) ----------
// per WG: b = blockIdx/SPLIT_, rows [seg*128, seg*128+128), double-buffered
// 8-row tiles fetched with GLOBAL_LOAD_ASYNC_TO_LDS_B128 when available.
//  pass1 (VALU): raw[h][n] = qk[b,h] . key[b,n] + qbk[b,h]
//  pass2 (WMMA f32 16x16x4): s_key[h][e] += raw[h][n] * key[n][e]
#define KPAD  1028            // padded row stride (floats): bank-spread + 16B aligned
#define TROWS 8               // key rows per tile
#define NTILES ((N_ / SPLIT_) / TROWS)   // 16

__device__ __forceinline__ void load_tile(const float* src, float* dst, int t) {
#if ASYNC_TILES
#pragma unroll
    for (int k = 0; k < TROWS; ++k) {
        __builtin_amdgcn_global_load_async_to_lds_b128(
            (__attribute__((address_space(1))) vi4*)(src + k * E_ + t * 4),
            (__attribute__((address_space(3))) vi4*)(dst + k * KPAD + t * 4),
            0, 0);
    }
#else
#pragma unroll
    for (int k = 0; k < TROWS; ++k) {
        float4 v = ((const float4*)(src + k * E_))[t];
        *(float4*)(dst + k * KPAD + t * 4) = v;
    }
#endif
}

__global__ __launch_bounds__(256) void attn_stream(
        const float* __restrict__ key, const float* __restrict__ qk,
        const float* __restrict__ qbk, float* __restrict__ s_key,
        float* __restrict__ sumraw) {
    __shared__ float qkT[E_ * H_];            // [e][h] transposed, 32 KB
    __shared__ float keyL[2][TROWS * KPAD];   // double-buffered 8-row key tiles
    __shared__ float scoresL[16 * 8];         // [h][n], rows h=8..15 stay zero (WMMA pad)
    __shared__ float partL[256 * 8];          // pass1 partials [t][h]
    __shared__ float qbkL[8];

    int t = threadIdx.x;
    int b = blockIdx.x / SPLIT_;
    int seg = blockIdx.x % SPLIT_;
    int wave = t >> 5, lane = t & 31;

    const float* qkb = qk + b * (H_ * E_);
    for (int i = t; i < H_ * E_; i += 256) {
        int h = i >> 10, e = i & 1023;
        qkT[e * 8 + h] = qkb[i];
    }
    if (t < 8) qbkL[t] = qbk[b * 8 + t];
    if (t >= 64 && t < 128) scoresL[t] = 0.f;   // zero pad rows h=8..15 (kept zero)

    v8f c[8] = {};                               // per-wave s_key accumulators: 8 e-tiles

    const float* keyb = key + ((size_t)b * N_ + (size_t)seg * (N_ / SPLIT_)) * E_;
    load_tile(keyb, keyL[0], t);                 // prologue: tile 0 in flight
    __syncthreads();

    for (int tile = 0; tile < NTILES; ++tile) {
        int cur = tile & 1;
        if (tile + 1 < NTILES)                   // overlap next tile's HBM traffic
            load_tile(keyb + (size_t)(tile + 1) * TROWS * E_, keyL[1 - cur], t);
#if ASYNC_TILES
        // async loads retire in order: <=TROWS outstanding => current tile complete
        if (tile + 1 < NTILES) __builtin_amdgcn_s_wait_asynccnt(TROWS);
        else                   __builtin_amdgcn_s_wait_asynccnt(0);
#else
        if (tile + 1 < NTILES)                   // gfx1250 global_prefetch_b8
            __builtin_prefetch(keyb + (size_t)(tile + 1) * TROWS * E_ + t * 32, 0, 0);
#endif
        __syncthreads();

        // ---- pass1: 8 score dots per row, register-blocked over h ----
        {
            int n = t & 7, s = t >> 3;           // 32 e-segments of 32
            float acc[8];
#pragma unroll
            for (int h = 0; h < 8; ++h) acc[h] = 0.f;
            const float*  kr = &keyL[cur][n * KPAD + s * 32];
            const float4* q4 = (const float4*)&qkT[s * 32 * 8];
            for (int e = 0; e < 32; ++e) {
                float  kv = kr[e];
                float4 qa = q4[e * 2], qb = q4[e * 2 + 1];
                acc[0] += kv * qa.x; acc[1] += kv * qa.y;
                acc[2] += kv * qa.z; acc[3] += kv * qa.w;
                acc[4] += kv * qb.x; acc[5] += kv * qb.y;
                acc[6] += kv * qb.z; acc[7] += kv * qb.w;
            }
#pragma unroll
            for (int h = 0; h < 8; ++h) partL[t * 8 + h] = acc[h];
        }
        __syncthreads();
        if (t < 64) {
            int n = t & 7, h = t >> 3;
            float s = qbkL[h];
            for (int sg = 0; sg < 32; ++sg) s += partL[(sg * 8 + n) * 8 + h];
            scoresL[h * 8 + n] = s;
        }
        __syncthreads();
        if (t < 8) {                              // sum of raw scores (for bv term)
            float s = 0.f;
            for (int n = 0; n < 8; ++n) s += scoresL[t * 8 + n];
            atomicAdd(&sumraw[b * 8 + t], s);
        }

        // ---- pass2: s_key tile accumulate via V_WMMA_F32_16X16X4_F32 ----
        // A(16x4)=scores rows h (8..15 zero), K=n; B(4x16)=key rows n, cols e.
        {
            int hi = lane >> 4;                   // lane half selects K pair
            int lo = lane & 15;
#pragma unroll
            for (int ks = 0; ks < 2; ++ks) {
                v2f a = *(const v2f*)&scoresL[lo * 8 + ks * 4 + 2 * hi];
#pragma unroll
                for (int j = 0; j < 8; ++j) {
                    int ebase = wave * 128 + j * 16;
                    int r0 = ks * 4 + 2 * hi;
                    v2f bm;
                    bm.x = keyL[cur][r0 * KPAD + ebase + lo];
                    bm.y = keyL[cur][(r0 + 1) * KPAD + ebase + lo];
                    c[j] = __builtin_amdgcn_wmma_f32_16x16x4_f32(
                        false, a, false, bm, (short)0, c[j], false, false);
                }
            }
        }
        __syncthreads();   // keyL[cur]/scoresL/partL reused next iteration
    }

    // writeback: C/D layout -> vgpr r, lanes 0-15 hold (h=r, e=lane); pad rows dropped
    if ((t & 31) < 16) {
        int lo = t & 31;
#pragma unroll
        for (int j = 0; j < 8; ++j) {
            int e = wave * 128 + j * 16 + lo;
#pragma unroll
            for (int r = 0; r < 8; ++r)
                atomicAdd(&s_key[((size_t)b * H_ + r) * E_ + e], c[j][r]);
        }
    }
}

// masked sum of value2 over n (one pass over 128 MB)
__global__ void v2sum(const float* __restrict__ value2, const int* __restrict__ mask,
                      float* __restrict__ vsum) {
    int b = blockIdx.x / VS_, seg = blockIdx.x % VS_;
    int t = threadIdx.x;                 // 256 threads * float4 = one row
    const int rows = N_ / VS_;           // 64
    const float* vb = value2 + ((size_t)b * N_ + (size_t)seg * rows) * E_;
    const int*   mb = mask + b * N_ + seg * rows;
    float a0 = 0.f, a1 = 0.f, a2 = 0.f, a3 = 0.f;
    for (int n = 0; n < rows; ++n) {
        if (mb[n] != 0) {
            float4 v = ((const float4*)(vb + (size_t)n * E_))[t];
            a0 += v.x; a1 += v.y; a2 += v.z; a3 += v.w;
        }
    }
    float* dst = vsum + b * E_ + t * 4;
    atomicAdd(dst + 0, a0); atomicAdd(dst + 1, a1);
    atomicAdd(dst + 2, a2); atomicAdd(dst + 3, a3);
}

// per (b,h) epilogue: attended -> hvec -> alpha_channel; v2_att from masked mean
__global__ __launch_bounds__(128) void finalize(
        const float* __restrict__ s_key, const float* __restrict__ sumraw,
        const float* __restrict__ vsum, const int* __restrict__ mask,
        const float* __restrict__ Wv, const float* __restrict__ bv,
        const float* __restrict__ Wb, const float* __restrict__ bb,
        const float* __restrict__ Wl2, const float* __restrict__ bl2,
        const float* __restrict__ value1, float* __restrict__ out) {
    __shared__ float attL[D_];
    __shared__ float hL[MID_];
    __shared__ float redL[D_];
    int b = blockIdx.x >> 3, h = blockIdx.x & 7;
    int d = threadIdx.x;

    float cm = 0.f;                                  // mask count
    for (int i = 0; i < N_ / D_; ++i) cm += (float)mask[b * N_ + i * D_ + d];
    redL[d] = cm;
    __syncthreads();
    for (int s = 64; s > 0; s >>= 1) {
        if (d < s) redL[d] += redL[d + s];
        __syncthreads();
    }
    float cnt = redL[0];

    int row = h * D_ + d;
    const float* wv = Wv + (size_t)row * E_;
    const float* sk = s_key + ((size_t)b * H_ + h) * E_;
    const float* vs = vsum + b * E_;
    float accA = 0.f, accV = 0.f;
    for (int e = 0; e < E_; ++e) { float w = wv[e]; accA += w * sk[e]; accV += w * vs[e]; }
    float att = 0.08838834764831845f * (accA + sumraw[b * H_ + h] * bv[row]); // 1/sqrt(128)
    attL[d] = att;
    __syncthreads();
    if (d < MID_) {
        const float* wb = Wb + d * D_;
        float a = bb[d];
        for (int i = 0; i < D_; ++i) a += wb[i] * attL[i];
        hL[d] = a > 0.f ? a : 0.f;
    }
    __syncthreads();
    const float* wl2 = Wl2 + d * MID_;
    float a = bl2[d];
    for (int m = 0; m < MID_; ++m) a += wl2[m] * hL[m];
    float ach = 1.f / (1.f + expf(-a));
    float v2a = accV / cnt + bv[row];                // uniform softmax over masked-in n
    out[b * E_ + row] = value1[b * E_ + row] * v2a * ach;
}

extern "C" void kernel_launch(void* const* d_in, const int* in_sizes, int n_in,
                              void* d_out, int out_size, void* d_ws, size_t ws_size,
                              hipStream_t stream) {
    const float* query  = (const float*)d_in[0];
    const float* key    = (const float*)d_in[1];
    const int*   mask   = (const int*)  d_in[2];
    const float* value1 = (const float*)d_in[3];
    const float* value2 = (const float*)d_in[4];
    const float* Wq = (const float*)d_in[5];
    const float* bq = (const float*)d_in[6];
    const float* Wk = (const float*)d_in[7];
    const float* bk = (const float*)d_in[8];
    const float* Wv = (const float*)d_in[9];
    const float* bv = (const float*)d_in[10];
    const float* Wb = (const float*)d_in[11];
    const float* bb = (const float*)d_in[12];
    // d_in[13]=Wl, d_in[14]=bl: algebraically irrelevant (softmax of a constant)
    const float* Wl2 = (const float*)d_in[15];
    const float* bl2 = (const float*)d_in[16];

    float* ws     = (float*)d_ws;
    float* Q      = ws + OFF_Q;
    float* qk     = ws + OFF_QK;
    float* qbk    = ws + OFF_QBK;
    float* s_key  = ws + OFF_SKEY;
    float* sumraw = ws + OFF_SUMR;
    float* vsum   = ws + OFF_VSUM;
    float* out    = (float*)d_out;

    zero_ws<<<(ZERO_COUNT + 255) / 256, 256, 0, stream>>>(ws);
    q_proj  <<<(B_ * E_) / 256, 256, 0, stream>>>(query, Wq, bq, Q);
    qk_proj <<<(B_ * H_ * E_) / 256, 256, 0, stream>>>(Q, Wk, qk);
    qbk_proj<<<1, 256, 0, stream>>>(Q, bk, qbk);
    attn_stream<<<B_ * SPLIT_, 256, 0, stream>>>(key, qk, qbk, s_key, sumraw);
    v2sum   <<<B_ * VS_, 256, 0, stream>>>(value2, mask, vsum);
    finalize<<<B_ * H_, 128, 0, stream>>>(s_key, sumraw, vsum, mask,
                                          Wv, bv, Wb, bb, Wl2, bl2, value1, out);
}